// RNN_Variational_Encoder_46153718563184
// MI455X (gfx1250) — compile-verified
//
#include <hip/hip_runtime.h>
#include <math.h>

// ---------------- problem constants ----------------
#define Bn 64
#define Tn 512
#define Hn 512
#define I0 80
#define K1 1024          // 2*H, layer-1 fan-in
#define LEAKc 0.9f

// scan partitioning
#define NWG 16           // workgroups per direction
#define COLS 32          // output columns per workgroup (NWG*COLS == Hn)
#define WSTR 516         // LDS row stride (floats), 516 % 64 == 4 -> conflict-free b64 reads

// gemm blocking
#define NTPW 4           // N tiles per wave (A-fragment register reuse)

typedef __attribute__((ext_vector_type(2))) float v2f;
typedef __attribute__((ext_vector_type(8))) float v8f;

// =====================================================================
// init: zero the h ping-pong buffers and the barrier counters
// =====================================================================
__global__ void init_ws_kernel(float* __restrict__ hbuf, int* __restrict__ cnt) {
    int i = blockIdx.x * blockDim.x + threadIdx.x;
    const int n = 4 * 2 * Bn * Hn;           // 4 (layer,dir) slots x double buffer
    if (i < n) hbuf[i] = 0.0f;
    if (i < 8) cnt[i] = 0;
}

// =====================================================================
// Input projection GEMM: xp[t][b][n] = sum_k A[b][t][k] * W[n][k]
// A is (B,T,K) row-major, W is (H,K) row-major, xp is (T,B,H).
// Each wave computes a 16x(16*NTPW) strip: one A fragment feeds NTPW WMMAs.
// =====================================================================
__global__ void gemm_xp_kernel(const float* __restrict__ A, int K,
                               const float* __restrict__ Wf,
                               const float* __restrict__ Wb,
                               float* __restrict__ xpf,
                               float* __restrict__ xpb) {
    const int lane = threadIdx.x & 31;
    const int wave = threadIdx.x >> 5;               // 8 waves / block
    const int NTG = (Hn / 16) / NTPW;                // 8 N-tile groups
    const int MT  = (Bn * Tn) / 16;                  // 2048 M tiles

    long g = (long)blockIdx.x * 8 + wave;            // global wave id
    int dir = (int)(g / (long)(MT * NTG));           // 0 = fwd, 1 = bwd weights
    int rem = (int)(g % (long)(MT * NTG));
    int mt = rem / NTG;
    int ng = rem % NTG;

    const float* W  = dir ? Wb : Wf;
    float*       xp = dir ? xpb : xpf;

    const int half = lane >> 4;                      // 0: K={0,1}, 1: K={2,3}
    const int lr   = lane & 15;
    const int koff = half * 2;

    // A fragment source row: m = t*B + b  ->  b = m % 64, t = m / 64
    const int m  = mt * 16 + lr;
    const int b  = m & (Bn - 1);
    const int t  = m >> 6;
    const float* arow = A + ((long)b * Tn + t) * K + koff;

    // B fragment base for N-tile group (tile q offset: +q*16*K rows)
    const float* brow0 = W + (long)(ng * NTPW * 16 + lr) * K + koff;

    v8f c[NTPW];
#pragma unroll
    for (int q = 0; q < NTPW; ++q) c[q] = (v8f){};

    for (int k0 = 0; k0 < K; k0 += 4) {
        v2f av = *(const v2f*)(arow + k0);
#pragma unroll
        for (int q = 0; q < NTPW; ++q) {
            v2f bv = *(const v2f*)(brow0 + (long)q * 16 * K + k0);
            c[q] = __builtin_amdgcn_wmma_f32_16x16x4_f32(false, av, false, bv,
                                                         (short)0, c[q],
                                                         false, false);
        }
    }

    // C layout: lane -> column (lane&15), VGPR r -> row r (+8 for lanes 16..31)
#pragma unroll
    for (int q = 0; q < NTPW; ++q) {
        const int n = (ng * NTPW + q) * 16 + lr;
        float* outp = xp + (long)(mt * 16 + half * 8) * Hn + n;
#pragma unroll
        for (int r = 0; r < 8; ++r)
            outp[(long)r * Hn] = c[q][r];
    }
}

// =====================================================================
// Recurrent scan for one layer (both directions concurrently).
// 2*NWG blocks of 256 threads. Each block owns COLS output columns of one
// direction; W slice lives in (dynamic) LDS; h state ping-pongs through
// global double buffers with an atomic-counter barrier each timestep.
// =====================================================================
__global__ void scan_kernel(const float* __restrict__ xpf,
                            const float* __restrict__ xpb,
                            const float* __restrict__ whh,   // (L,2,H,H)
                            const int*   __restrict__ lengths,
                            float* __restrict__ hbuf,        // 4 slots x 2 x (B*H)
                            float* __restrict__ hidden,      // (B,T,2H) or nullptr
                            float* __restrict__ out,         // (B, 4H)
                            int layer,
                            int* __restrict__ cnt) {
    extern __shared__ float wlds[];                  // COLS * WSTR floats

    const int dir    = blockIdx.x / NWG;             // 0 fwd, 1 bwd
    const int wg     = blockIdx.x % NWG;
    const int dirIdx = layer * 2 + dir;

    const float* W  = whh + (long)dirIdx * Hn * Hn;
    const float* xp = dir ? xpb : xpf;
    float* hb0 = hbuf + (long)dirIdx * 2 * (Bn * Hn);
    float* hb1 = hb0 + Bn * Hn;
    int* counter = cnt + dirIdx;

    // stage W[wg*COLS .. +COLS)[0..H) into LDS, padded stride
    for (int idx = threadIdx.x; idx < COLS * Hn; idx += blockDim.x) {
        int c = idx >> 9;                            // / 512
        int k = idx & (Hn - 1);
        wlds[c * WSTR + k] = W[(long)(wg * COLS + c) * Hn + k];
    }
    __syncthreads();

    const int lane = threadIdx.x & 31;
    const int wave = threadIdx.x >> 5;               // 8 waves
    const int mt    = wave >> 1;                     // batch tile 0..3
    const int ntile = wave & 1;                      // local col tile 0..1
    const int half  = lane >> 4;
    const int lr    = lane & 15;
    const int koff  = half * 2;

    const int rowA = mt * 16 + lr;                   // batch row for A fragment
    const int colL = ntile * 16 + lr;                // local column 0..31
    const int j    = wg * COLS + colL;               // global column
    const float* bld = wlds + colL * WSTR + koff;
    const int rbase = mt * 16 + half * 8;            // first C-fragment row

    // hoist per-lane row lengths (rows handled in the C fragment)
    int lenr[8];
#pragma unroll
    for (int r = 0; r < 8; ++r)
        lenr[r] = lengths[rbase + r];

    for (int s = 0; s < Tn; ++s) {
        const float* hcur  = (s & 1) ? hb1 : hb0;
        float*       hnext = (s & 1) ? hb0 : hb1;
        const float* arow  = hcur + (long)rowA * Hn + koff;
        const int tAct = dir ? (Tn - 1 - s) : s;

        // issue epilogue operands up-front so they overlap the WMMA chain
        float xpv[8], holdv[8];
#pragma unroll
        for (int r = 0; r < 8; ++r) {
            xpv[r]   = xp[((long)tAct * Bn + (rbase + r)) * Hn + j];
            holdv[r] = hcur[(long)(rbase + r) * Hn + j];
        }

        // prefetch next timestep's xp rows into WGP$ (global_prefetch_b8)
        const int tNext = dir ? (tAct - 1) : (tAct + 1);
        if (tNext >= 0 && tNext < Tn) {
#pragma unroll
            for (int r = 0; r < 8; ++r)
                __builtin_prefetch(xp + ((long)tNext * Bn + (rbase + r)) * Hn + j,
                                   0, 1);
        }

        v8f c = {};
        for (int k0 = 0; k0 < Hn; k0 += 4) {
            v2f av = *(const v2f*)(arow + k0);
            v2f bv = *(const v2f*)(bld + k0);
            c = __builtin_amdgcn_wmma_f32_16x16x4_f32(false, av, false, bv,
                                                      (short)0, c, false, false);
        }

#pragma unroll
        for (int r = 0; r < 8; ++r) {
            const int m = rbase + r;                 // batch index
            const bool msk = tAct < lenr[r];
            const float pre  = c[r] + xpv[r];
            const float upd  = (1.0f - LEAKc) * holdv[r] + LEAKc * tanhf(pre);
            const float hnew = msk ? upd : holdv[r];
            hnext[(long)m * Hn + j] = hnew;
            if (hidden)
                hidden[((long)m * Tn + tAct) * (2 * Hn) + dir * Hn + j]
                    = msk ? hnew : 0.0f;
            if (s == Tn - 1)
                out[(long)m * (4 * Hn) + dirIdx * Hn + j] = hnew;
        }

        // grid barrier among the NWG blocks of this direction
        __syncthreads();
        if (threadIdx.x == 0) {
            __threadfence();                          // release our writes
            atomicAdd(counter, 1);
            const int target = NWG * (s + 1);
            while (__hip_atomic_load(counter, __ATOMIC_ACQUIRE,
                                     __HIP_MEMORY_SCOPE_AGENT) < target) {
                __builtin_amdgcn_s_sleep(2);
            }
            __threadfence();                          // acquire others' writes
        }
        __syncthreads();
    }
}

// =====================================================================
// host-side orchestration
// =====================================================================
extern "C" void kernel_launch(void* const* d_in, const int* in_sizes, int n_in,
                              void* d_out, int out_size, void* d_ws, size_t ws_size,
                              hipStream_t stream) {
    const float* x       = (const float*)d_in[0];
    const int*   lengths = (const int*)  d_in[1];
    const float* w_ih0_f = (const float*)d_in[2];
    const float* w_ih0_b = (const float*)d_in[3];
    const float* w_ih1_f = (const float*)d_in[4];
    const float* w_ih1_b = (const float*)d_in[5];
    const float* w_hh    = (const float*)d_in[6];
    float* out = (float*)d_out;

    char* ws = (char*)d_ws;
    const size_t XP_BYTES  = (size_t)Tn * Bn * Hn * 4;        // 64 MB
    const size_t HID_BYTES = (size_t)Bn * Tn * 2 * Hn * 4;    // 128 MB
    float* xpf  = (float*)(ws);
    float* xpb  = (float*)(ws + XP_BYTES);
    float* hid  = (float*)(ws + 2 * XP_BYTES);
    float* hbuf = (float*)(ws + 2 * XP_BYTES + HID_BYTES);
    int*   cnt  = (int*)  (ws + 2 * XP_BYTES + HID_BYTES
                              + (size_t)4 * 2 * Bn * Hn * 4);

    const int gemmBlocks =
        2 * ((Bn * Tn) / 16) * ((Hn / 16) / NTPW) / 8;        // 4096
    const size_t scanLds = (size_t)COLS * WSTR * sizeof(float);   // 66,048 B

    init_ws_kernel<<<1024, 256, 0, stream>>>(hbuf, cnt);

    // layer 0
    gemm_xp_kernel<<<gemmBlocks, 256, 0, stream>>>(x, I0, w_ih0_f, w_ih0_b, xpf, xpb);
    scan_kernel<<<2 * NWG, 256, scanLds, stream>>>(xpf, xpb, w_hh, lengths,
                                                   hbuf, hid, out, 0, cnt);
    // layer 1
    gemm_xp_kernel<<<gemmBlocks, 256, 0, stream>>>(hid, K1, w_ih1_f, w_ih1_b, xpf, xpb);
    scan_kernel<<<2 * NWG, 256, scanLds, stream>>>(xpf, xpb, w_hh, lengths,
                                                   hbuf, nullptr, out, 1, cnt);
}